// CrossAttention_70987219468874
// MI455X (gfx1250) — compile-verified
//
#include <hip/hip_runtime.h>

typedef __attribute__((ext_vector_type(2))) float v2f;
typedef __attribute__((ext_vector_type(8))) float v8f;

#define BB 8
#define CC 1024
#define LL 4096
#define CTXN 768
#define NG 8
#define CPG 128
#define EPSF 1e-5f
#define LSPLIT 8
#define LCHUNK (LL / LSPLIT)

// ws layout (float offsets)
#define O_CTXLN 0
#define O_KV    (O_CTXLN + BB * CTXN)        // 8 x 2048 : k = [0:1024), v = [1024:2048)
#define O_W     (O_KV + BB * 2 * CC)         // 8 x 1024
#define O_COEF  (O_W + BB * CC)              // 8 x 1024 (w * gn_gamma)
#define O_PV    (O_COEF + BB * CC)           // 8 x 1024 (Wp @ v)
#define O_P     (O_PV + BB * CC)             // 8 x 8 x 4096 partial group dots
#define O_PS    (O_P + BB * NG * LL)         // 64 x 8 partial sums
#define O_PS2   (O_PS + BB * NG * LSPLIT)    // 64 x 8 partial sumsq
#define O_ATTN  (O_PS2 + BB * NG * LSPLIT)   // 8 x 4096

// ---------------- A1: LayerNorm of context (B rows of 768) ----------------
__global__ __launch_bounds__(256) void kA1_ln(const float* __restrict__ ctx,
                                              const float* __restrict__ lng,
                                              const float* __restrict__ lnb,
                                              float* __restrict__ out) {
    int b = blockIdx.x, t = threadIdx.x;
    __shared__ float rs[256], rq[256];
    float s = 0.f, q = 0.f;
    for (int i = t; i < CTXN; i += 256) {
        float v = ctx[b * CTXN + i];
        s += v; q += v * v;
    }
    rs[t] = s; rq[t] = q; __syncthreads();
    for (int off = 128; off; off >>= 1) {
        if (t < off) { rs[t] += rs[t + off]; rq[t] += rq[t + off]; }
        __syncthreads();
    }
    float mu = rs[0] / (float)CTXN;
    float var = rq[0] / (float)CTXN - mu * mu;
    float rstd = rsqrtf(var + EPSF);
    for (int i = t; i < CTXN; i += 256)
        out[b * CTXN + i] = (ctx[b * CTXN + i] - mu) * rstd * lng[i] + lnb[i];
}

// ---------------- A2: kv = ctx_ln @ Wkv^T + bkv  (8 x 2048) ----------------
__global__ __launch_bounds__(256) void kA2_kv(const float* __restrict__ ctxln,
                                              const float* __restrict__ Wkv,
                                              const float* __restrict__ bkv,
                                              float* __restrict__ kv) {
    int tid = blockIdx.x * 256 + threadIdx.x;  // 16384 threads
    int b = tid / (2 * CC), j = tid % (2 * CC);
    const float* cr = ctxln + b * CTXN;
    const float* wr = Wkv + (size_t)j * CTXN;
    float acc = bkv[j];
    for (int t = 0; t < CTXN; ++t) acc += cr[t] * wr[t];
    kv[b * 2 * CC + j] = acc;
}

// ------- A3: w = k @ Wq, coef = w*gamma ; pv = Wp @ v  (both 8 x 1024) -------
__global__ __launch_bounds__(256) void kA3_wpv(const float* __restrict__ kv,
                                               const float* __restrict__ Wq,
                                               const float* __restrict__ Wp,
                                               const float* __restrict__ gng,
                                               float* __restrict__ w,
                                               float* __restrict__ coef,
                                               float* __restrict__ pv) {
    int tid = blockIdx.x * 256 + threadIdx.x;  // 2*8192 threads
    int half = tid / (BB * CC);
    int r = tid % (BB * CC);
    int b = r / CC, c = r % CC;
    if (half == 0) {
        // w[b,c] = sum_o k[b,o] * Wq[o,c]   (coalesced over c)
        const float* kb = kv + b * 2 * CC;
        float acc = 0.f;
        for (int o = 0; o < CC; ++o) acc += kb[o] * Wq[(size_t)o * CC + c];
        w[b * CC + c] = acc;
        coef[b * CC + c] = acc * gng[c];
    } else {
        // pv[b,c] = sum_cc Wp[c,cc] * v[b,cc]
        const float* vb = kv + b * 2 * CC + CC;
        const float* wr = Wp + (size_t)c * CC;
        float acc = 0.f;
        for (int cc = 0; cc < CC; ++cc) acc += wr[cc] * vb[cc];
        pv[b * CC + c] = acc;
    }
}

// ---- B: fused GroupNorm-stats + group-dot pass over x, via WMMA f32 16x16x4 ----
// A-matrix row0 = coef (dot), row1 = 1.0 (column sums); VALU squares -> sumsq.
// A-fragment built with unconditional LDS reads + cndmask (no divergent exec).
__global__ __launch_bounds__(256) void kB_wmma(const float* __restrict__ x,
                                               const float* __restrict__ coef,
                                               float* __restrict__ p,
                                               float* __restrict__ partSum,
                                               float* __restrict__ partSumSq) {
    int blk = blockIdx.x;              // B*NG*LSPLIT = 512
    int split = blk % LSPLIT;
    int bg = blk / LSPLIT;
    int g = bg % NG, b = bg / NG;
    int lane = threadIdx.x & 31, wave = threadIdx.x >> 5;

    __shared__ float sCoef[CPG];
    __shared__ float sRedS[8], sRedQ[8];
    if (threadIdx.x < CPG) sCoef[threadIdx.x] = coef[b * CC + g * CPG + threadIdx.x];
    __syncthreads();

    const int m = lane & 15;       // M index for A, N index for B/D
    const int h = lane >> 4;       // K half
    const bool is0 = (m == 0);     // hoisted lane predicates -> single v_cmp
    const float sel1 = (m == 1) ? 1.f : 0.f;   // ones-row value (loop invariant)
    int lbase = split * LCHUNK;
    float sumAcc = 0.f, sqAcc = 0.f;

    for (int t = 0; t < LCHUNK / (16 * 8); ++t) {   // 4 tiles per wave
        int l0 = lbase + (wave + 8 * t) * 16;
        const float* xb = x + ((size_t)b * CC + (size_t)g * CPG) * LL + l0 + m;
        v8f acc = {0.f, 0.f, 0.f, 0.f, 0.f, 0.f, 0.f, 0.f};
#pragma unroll
        for (int kk = 0; kk < CPG; kk += 4) {
            int k0 = kk + 2 * h;
            float b0 = xb[(size_t)k0 * LL];
            float b1 = xb[(size_t)(k0 + 1) * LL];
            // unconditional LDS reads; unused lanes' values discarded by cndmask
            float c0 = sCoef[k0];
            float c1 = sCoef[k0 + 1];
            v2f bf; bf[0] = b0; bf[1] = b1;
            v2f af;
            af[0] = is0 ? c0 : sel1;
            af[1] = is0 ? c1 : sel1;
            acc = __builtin_amdgcn_wmma_f32_16x16x4_f32(
                false, af, false, bf, (short)0, acc, false, false);
            sqAcc += b0 * b0 + b1 * b1;
        }
        if (lane < 16) {
            // D row M=0 (dot) lives in acc[0] lanes 0-15; row M=1 (colsum) in acc[1]
            p[((size_t)b * NG + g) * LL + l0 + lane] = acc[0];
            sumAcc += acc[1];
        }
    }
    for (int off = 16; off; off >>= 1) {
        sumAcc += __shfl_down(sumAcc, off, 32);
        sqAcc  += __shfl_down(sqAcc,  off, 32);
    }
    if (lane == 0) { sRedS[wave] = sumAcc; sRedQ[wave] = sqAcc; }
    __syncthreads();
    if (threadIdx.x == 0) {
        float s = 0.f, q = 0.f;
        for (int i = 0; i < 8; ++i) { s += sRedS[i]; q += sRedQ[i]; }
        partSum[bg * LSPLIT + split] = s;
        partSumSq[bg * LSPLIT + split] = q;
    }
}

// ---------------- C: finalize stats, logits, softmax per batch ----------------
__global__ __launch_bounds__(256) void kC_softmax(const float* __restrict__ w,
                                                  const float* __restrict__ coef,
                                                  const float* __restrict__ kv,
                                                  const float* __restrict__ bq,
                                                  const float* __restrict__ gnb,
                                                  const float* __restrict__ partSum,
                                                  const float* __restrict__ partSumSq,
                                                  const float* __restrict__ p,
                                                  float* __restrict__ attn) {
    int b = blockIdx.x, t = threadIdx.x;
    __shared__ float sMu[NG], sRsig[NG], sSg[NG];
    __shared__ float rA[256], rB[256];
    __shared__ float sConst, sMax, sSum;

    if (t < NG) {
        float s = 0.f, q = 0.f;
        for (int i = 0; i < LSPLIT; ++i) {
            s += partSum[(b * NG + t) * LSPLIT + i];
            q += partSumSq[(b * NG + t) * LSPLIT + i];
        }
        float invN = 1.f / (float)(CPG * LL);
        float mu = s * invN;
        float var = q * invN - mu * mu;
        sMu[t] = mu;
        sRsig[t] = rsqrtf(var + EPSF);
        sSg[t] = 0.f;
    }
    __syncthreads();

    float wbeta = 0.f, kbq = 0.f;
    for (int c = t; c < CC; c += 256) {
        wbeta += w[b * CC + c] * gnb[c];
        kbq += kv[b * 2 * CC + c] * bq[c];
        atomicAdd(&sSg[c / CPG], coef[b * CC + c]);
    }
    rA[t] = wbeta; rB[t] = kbq; __syncthreads();
    for (int off = 128; off; off >>= 1) {
        if (t < off) { rA[t] += rA[t + off]; rB[t] += rB[t + off]; }
        __syncthreads();
    }
    if (t == 0) {
        float cst = rA[0] + rB[0];
        for (int g = 0; g < NG; ++g) cst -= sRsig[g] * sMu[g] * sSg[g];
        sConst = cst;
    }
    __syncthreads();

    const float scale = rsqrtf((float)CC);  // HEAD_DIM^-0.5
    float z[16];
    float lmax = -3.0e38f;
    for (int i = 0; i < 16; ++i) {
        int l = i * 256 + t;
        float acc = sConst;
        for (int g = 0; g < NG; ++g)
            acc += sRsig[g] * p[((size_t)b * NG + g) * LL + l];
        z[i] = acc * scale;
        lmax = fmaxf(lmax, z[i]);
    }
    rA[t] = lmax; __syncthreads();
    for (int off = 128; off; off >>= 1) {
        if (t < off) rA[t] = fmaxf(rA[t], rA[t + off]);
        __syncthreads();
    }
    if (t == 0) sMax = rA[0];
    __syncthreads();
    float lsum = 0.f;
    for (int i = 0; i < 16; ++i) {
        float e = __expf(z[i] - sMax);
        z[i] = e; lsum += e;
    }
    rA[t] = lsum; __syncthreads();
    for (int off = 128; off; off >>= 1) {
        if (t < off) rA[t] += rA[t + off];
        __syncthreads();
    }
    if (t == 0) sSum = rA[0];
    __syncthreads();
    float inv = 1.f / sSum;
    for (int i = 0; i < 16; ++i)
        attn[b * LL + i * 256 + t] = z[i] * inv;
}

// ------------- D: out = x + pv[b,c] * attn[b,l] + bp[c]  (float4) -------------
__global__ __launch_bounds__(256) void kD_epilogue(const float4* __restrict__ x4,
                                                   const float* __restrict__ pv,
                                                   const float* __restrict__ bp,
                                                   const float* __restrict__ attn,
                                                   float4* __restrict__ out4) {
    int i = blockIdx.x * 256 + threadIdx.x;   // B*C*L/4 elements, exact grid
    const int L4 = LL / 4;
    int l4 = i % L4;
    int rem = i / L4;
    int c = rem % CC, b = rem / CC;
    float4 xv = x4[i];
    float4 av = *(const float4*)(attn + b * LL + 4 * l4);
    float s = pv[b * CC + c];
    float bb = bp[c];
    float4 o;
    o.x = xv.x + s * av.x + bb;
    o.y = xv.y + s * av.y + bb;
    o.z = xv.z + s * av.z + bb;
    o.w = xv.w + s * av.w + bb;
    out4[i] = o;
}

extern "C" void kernel_launch(void* const* d_in, const int* in_sizes, int n_in,
                              void* d_out, int out_size, void* d_ws, size_t ws_size,
                              hipStream_t stream) {
    const float* x    = (const float*)d_in[0];
    const float* ctx  = (const float*)d_in[1];
    const float* gng  = (const float*)d_in[2];
    const float* gnb  = (const float*)d_in[3];
    const float* lng  = (const float*)d_in[4];
    const float* lnb  = (const float*)d_in[5];
    const float* Wq   = (const float*)d_in[6];
    const float* bq   = (const float*)d_in[7];
    const float* Wkv  = (const float*)d_in[8];
    const float* bkv  = (const float*)d_in[9];
    const float* Wp   = (const float*)d_in[10];
    const float* bp   = (const float*)d_in[11];
    float* out = (float*)d_out;
    float* ws  = (float*)d_ws;

    float* ctxln = ws + O_CTXLN;
    float* kv    = ws + O_KV;
    float* w     = ws + O_W;
    float* coef  = ws + O_COEF;
    float* pv    = ws + O_PV;
    float* p     = ws + O_P;
    float* ps    = ws + O_PS;
    float* ps2   = ws + O_PS2;
    float* attn  = ws + O_ATTN;

    kA1_ln<<<BB, 256, 0, stream>>>(ctx, lng, lnb, ctxln);
    kA2_kv<<<(BB * 2 * CC) / 256, 256, 0, stream>>>(ctxln, Wkv, bkv, kv);
    kA3_wpv<<<(2 * BB * CC) / 256, 256, 0, stream>>>(kv, Wq, Wp, gng, w, coef, pv);
    kB_wmma<<<BB * NG * LSPLIT, 256, 0, stream>>>(x, coef, p, ps, ps2);
    kC_softmax<<<BB, 256, 0, stream>>>(w, coef, kv, bq, gnb, ps, ps2, p, attn);
    kD_epilogue<<<(BB * CC * LL / 4) / 256, 256, 0, stream>>>(
        (const float4*)x, pv, bp, attn, (float4*)out);
}